// Inception_Block4D_8538394985139
// MI455X (gfx1250) — compile-verified
//
#include <hip/hip_runtime.h>

typedef float v2f __attribute__((ext_vector_type(2)));
typedef float v8f __attribute__((ext_vector_type(8)));

#define XSTRIDE 18                    // floats per spatial position in LDS (pad: 8B aligned, few bank conflicts)
#define ZP_IDX  (4096 * XSTRIDE)      // zero-page float index (for out-of-bounds taps)
#define LDS_FLOATS (ZP_IDX + 16)

struct WPtrs { const float* p[6]; };

// Re-tile weights [o][c][tap] -> [tap][o][c] so WMMA A-fragments load as contiguous b64.
__global__ void transposeW(const float* __restrict__ w, float* __restrict__ wt, int k4) {
    int idx = blockIdx.x * blockDim.x + threadIdx.x;
    int total = k4 * 256;
    if (idx >= total) return;
    int tap = idx >> 8;
    int oc  = idx & 255;              // o*16 + c
    wt[idx] = w[oc * k4 + tap];
}

template <bool TRANS>
__global__ __launch_bounds__(64) void conv4d_wmma(const float* __restrict__ x,
                                                  float* __restrict__ out,
                                                  WPtrs wp) {
    extern __shared__ float xlds[];
    const int tid = threadIdx.x;

    // Stage x transposed: xlds[s*XSTRIDE + c] = x[c*4096 + s]  (coalesced global reads)
    for (int i = tid; i < 65536; i += 64) {
        int c = i >> 12;
        int s = i & 4095;
        xlds[s * XSTRIDE + c] = x[i];
    }
    if (tid < 16) xlds[ZP_IDX + tid] = 0.0f;   // zero page for clipped taps
    __syncthreads();

    const int wave = tid >> 5;
    const int lane = tid & 31;
    const int n    = lane & 15;       // M (A) / N (B,C) index
    const int hi   = lane >> 4;       // lane half selects K pair {0,1} vs {2,3}
    const int hi2  = hi * 2;
    const int dd3  = n >> 3;          // row within tile (d3 offset 0/1)
    const int d4   = n & 7;

    const int tileId = blockIdx.x * 2 + wave;        // 128 blocks * 2 waves = 256 tiles
    const int d1 = tileId >> 5;
    const int d2 = (tileId >> 2) & 7;
    const int D3 = (tileId & 3) * 2;

    const int aoff = n * 16 + hi2;    // per-lane float offset inside a 256-float tap block

    v8f acc = {0.f, 0.f, 0.f, 0.f, 0.f, 0.f, 0.f, 0.f};

    for (int ki = 0; ki < 6; ++ki) {
        const int k  = 2 * ki + 1;
        const int p  = ki;
        const int k2 = k * k;
        const int k4 = k2 * k2;
        const float* __restrict__ w = wp.p[ki];
        int rowIdx = 0;
        if (!TRANS) rowIdx = (n * 16 + hi2) * k4;   // strided fallback layout

        for (int t1 = 0; t1 < k; ++t1) {
            const int c1 = d1 + t1 - p;
            if ((unsigned)c1 >= 8u) continue;       // scalar skip (fully clipped)
            for (int t2 = 0; t2 < k; ++t2) {
                const int c2 = d2 + t2 - p;
                if ((unsigned)c2 >= 8u) continue;
                const int base12 = (c1 * 8 + c2) * 64;
                const int tap12  = (t1 * k + t2) * k2;
                for (int t3 = 0; t3 < k; ++t3) {
                    const int c3b = D3 + t3 - p;
                    if (c3b < -1 || c3b > 7) continue;     // both tile rows clipped
                    const int  c3 = c3b + dd3;             // per-lane coord
                    const bool m3 = (unsigned)c3 < 8u;
                    const int  idxBase = (base12 + c3 * 8) * XSTRIDE + hi2; // garbage if !m3 (muxed away)
                    const int  tap3 = tap12 + t3 * k;
                    for (int t4 = 0; t4 < k; ++t4) {
                        const int  c4  = d4 + t4 - p;
                        const bool ok  = m3 & ((unsigned)c4 < 8u);
                        const int  bix = ok ? (idxBase + c4 * XSTRIDE) : ZP_IDX;
                        const float* bp = &xlds[bix];
                        v2f b0 = *(const v2f*)(bp);
                        v2f b1 = *(const v2f*)(bp + 4);
                        v2f b2 = *(const v2f*)(bp + 8);
                        v2f b3 = *(const v2f*)(bp + 12);

                        const int tap = tap3 + t4;
                        v2f a0, a1, a2, a3;
                        if (TRANS) {
                            const float* ap = w + tap * 256 + aoff;
                            a0 = *(const v2f*)(ap);
                            a1 = *(const v2f*)(ap + 4);
                            a2 = *(const v2f*)(ap + 8);
                            a3 = *(const v2f*)(ap + 12);
                        } else {
                            const float* ap = w + rowIdx + tap;
                            a0.x = ap[0];           a0.y = ap[k4];
                            a1.x = ap[4 * k4];      a1.y = ap[5 * k4];
                            a2.x = ap[8 * k4];      a2.y = ap[9 * k4];
                            a3.x = ap[12 * k4];     a3.y = ap[13 * k4];
                        }
                        // D = A(16x4) * B(4x16) + C, chained over K=16 channels
                        acc = __builtin_amdgcn_wmma_f32_16x16x4_f32(false, a0, false, b0, (short)0, acc, false, false);
                        acc = __builtin_amdgcn_wmma_f32_16x16x4_f32(false, a1, false, b1, (short)0, acc, false, false);
                        acc = __builtin_amdgcn_wmma_f32_16x16x4_f32(false, a2, false, b2, (short)0, acc, false, false);
                        acc = __builtin_amdgcn_wmma_f32_16x16x4_f32(false, a3, false, b3, (short)0, acc, false, false);
                    }
                }
            }
        }
    }

    // C/D layout: VGPR v holds row M = v + hi*8, column N = n
    const float sc = 1.0f / 6.0f;
    float* op = out + (hi * 8) * 4096 + tileId * 16 + n;
#pragma unroll
    for (int v = 0; v < 8; ++v)
        op[v * 4096] = acc[v] * sc;
}

extern "C" void kernel_launch(void* const* d_in, const int* in_sizes, int n_in,
                              void* d_out, int out_size, void* d_ws, size_t ws_size,
                              hipStream_t stream) {
    (void)in_sizes; (void)n_in; (void)out_size;
    const float* x = (const float*)d_in[0];
    float* out = (float*)d_out;

    int k4[6]; size_t off[6]; size_t tot = 0;
    for (int i = 0; i < 6; ++i) {
        int k = 2 * i + 1;
        k4[i] = k * k * k * k;
        off[i] = tot;
        tot += (size_t)k4[i] * 256;
    }
    const bool trans = (ws_size >= tot * sizeof(float));

    WPtrs wp;
    if (trans) {
        float* wt = (float*)d_ws;
        for (int i = 0; i < 6; ++i) {
            int total = k4[i] * 256;
            transposeW<<<(total + 255) / 256, 256, 0, stream>>>(
                (const float*)d_in[1 + i], wt + off[i], k4[i]);
            wp.p[i] = wt + off[i];
        }
    } else {
        for (int i = 0; i < 6; ++i) wp.p[i] = (const float*)d_in[1 + i];
    }

    const size_t lds_bytes = (size_t)LDS_FLOATS * sizeof(float);
    if (trans)
        conv4d_wmma<true><<<128, 64, lds_bytes, stream>>>(x, out, wp);
    else
        conv4d_wmma<false><<<128, 64, lds_bytes, stream>>>(x, out, wp);
}